// Hierarchical_encoder_58102317580559
// MI455X (gfx1250) — compile-verified
//
#include <hip/hip_runtime.h>
#include <hip/hip_bf16.h>

#define D 128
#define MAXS 20

typedef __attribute__((ext_vector_type(2))) float v2f;
typedef __attribute__((ext_vector_type(8))) float v8f;

__device__ __forceinline__ float wave_sum(float v) {
#pragma unroll
    for (int off = 16; off > 0; off >>= 1)
        v += __shfl_xor(v, off, 32);
    return v;
}

// ---------------------------------------------------------------------------
// Intra-view attention: one wave32 per node. Gather S neighbor rows (float4 /
// lane -> global_load_b128), stage them in LDS (ds_store_b128), score with
// wave-shuffle dot-product reductions, softmax(leaky_relu), weighted sum, ELU.
// ---------------------------------------------------------------------------
__global__ void intra_kernel(const float* __restrict__ h_ref,
                             const float* __restrict__ h,
                             const int*   __restrict__ nei,
                             const float* __restrict__ att,   // 2*D
                             float*       __restrict__ e_out, // N*D
                             int N, int S)
{
    __shared__ float rows[4][MAXS * D];
    __shared__ float scr[4][MAXS];

    const int wave = threadIdx.x >> 5;   // 4 waves / block, 1 node / wave
    const int lane = threadIdx.x & 31;
    const int node = blockIdx.x * 4 + wave;
    if (node >= N) return;
    if (S > MAXS) S = MAXS;

    const float4 ar = ((const float4*)att)[lane];          // att[:D]
    const float4 an = ((const float4*)(att + D))[lane];    // att[D:]
    const float4 hr = ((const float4*)(h_ref + (size_t)node * D))[lane];
    const float sref = wave_sum(hr.x * ar.x + hr.y * ar.y + hr.z * ar.z + hr.w * ar.w);

    const int* nrow = nei + (size_t)node * S;
    for (int s = 0; s < S; ++s) {
        const int idx = nrow[s];
        const float4 v = ((const float4*)(h + (size_t)idx * D))[lane];
        ((float4*)&rows[wave][s * D])[lane] = v;
        float sc = sref + wave_sum(v.x * an.x + v.y * an.y + v.z * an.z + v.w * an.w);
        sc = sc > 0.f ? sc : 0.01f * sc;                   // leaky_relu
        if (lane == 0) scr[wave][s] = sc;
    }

    float mx = -3.4e38f;
    for (int s = 0; s < S; ++s) mx = fmaxf(mx, scr[wave][s]);

    float den = 0.f;
    float4 acc = make_float4(0.f, 0.f, 0.f, 0.f);
    for (int s = 0; s < S; ++s) {
        const float w = __expf(scr[wave][s] - mx);
        den += w;
        const float4 v = ((const float4*)&rows[wave][s * D])[lane];
        acc.x += w * v.x; acc.y += w * v.y; acc.z += w * v.z; acc.w += w * v.w;
    }
    const float inv = 1.f / den;
    acc.x *= inv; acc.y *= inv; acc.z *= inv; acc.w *= inv;

    // ELU (alpha = 1)
    acc.x = acc.x > 0.f ? acc.x : __expf(acc.x) - 1.f;
    acc.y = acc.y > 0.f ? acc.y : __expf(acc.y) - 1.f;
    acc.z = acc.z > 0.f ? acc.z : __expf(acc.z) - 1.f;
    acc.w = acc.w > 0.f ? acc.w : __expf(acc.w) - 1.f;

    ((float4*)(e_out + (size_t)node * D))[lane] = acc;
}

// ---------------------------------------------------------------------------
// FC column-sum: sps_sum[c] += sum_n tanh((e @ fcW^T)[n][c] + fcb[c]).
// 256 threads = 8 waves; wave w owns output columns [16w, 16w+16). Each wave
// computes 16x16 tiles with V_WMMA_F32_16X16X4_F32 over K=128 (32 WMMA ops),
// grid-striding over 16-row blocks. B tile (fcW^T slice) preloaded once.
//
// Layouts (ISA 7.12.2): A 16x4: lanes 0-15 -> M=lane, K={k0,k0+1}; lanes
// 16-31 -> K={k0+2,k0+3}. B 4x16 mirrored. C/D: lane<16 -> col=lane, rows
// 0-7 in v0..v7; lane>=16 -> col=lane-16, rows 8-15.
// ---------------------------------------------------------------------------
__global__ void fc_colsum_kernel(const float* __restrict__ e,    // N x D
                                 const float* __restrict__ fcW,  // D x D
                                 const float* __restrict__ fcb,  // D
                                 float*       __restrict__ sps_sum, // D accum
                                 int N)
{
    const int wave  = threadIdx.x >> 5;
    const int lane  = threadIdx.x & 31;
    const int col   = wave * 16 + (lane & 15);
    const int khalf = (lane >> 4) << 1;   // 0 or 2
    const int mrow0 = (lane >> 4) << 3;   // C rows 0..7 or 8..15

    // Preload B = fcW^T tile for this wave's 16 columns: B[k][n] = fcW[n][k].
    v2f B[32];
#pragma unroll
    for (int kk = 0; kk < 32; ++kk) {
        const float2 bv = *(const float2*)(fcW + (size_t)col * D + kk * 4 + khalf);
        B[kk].x = bv.x; B[kk].y = bv.y;
    }
    const float bcol = fcb[col];

    float colsum = 0.f;
    const int rtile = lane & 15;
    for (int rb = blockIdx.x * 16; rb < N; rb += gridDim.x * 16) {
        const int row  = rb + rtile;
        const int rowc = row < N ? row : N - 1;           // clamp addr; masked below
        const float* erow = e + (size_t)rowc * D;

        v8f c = {};
#pragma unroll
        for (int kk = 0; kk < 32; ++kk) {
            const float2 av = *(const float2*)(erow + kk * 4 + khalf);
            v2f a; a.x = av.x; a.y = av.y;
            c = __builtin_amdgcn_wmma_f32_16x16x4_f32(
                    false, a, false, B[kk], (short)0, c, false, false);
        }

#pragma unroll
        for (int i = 0; i < 8; ++i) {
            const int m = rb + mrow0 + i;
            colsum += (m < N) ? tanhf(c[i] + bcol) : 0.f;
        }
    }

    colsum += __shfl_xor(colsum, 16, 32);   // fold the two row-halves per column
    if (lane < 16) atomicAdd(&sps_sum[col], colsum);
}

// ---------------------------------------------------------------------------
__global__ void zero_kernel(float* p, int n)
{
    const int i = blockIdx.x * blockDim.x + threadIdx.x;
    if (i < n) p[i] = 0.f;
}

__global__ void beta_kernel(const float* __restrict__ sps,       // 2*D col sums
                            const float* __restrict__ att_inter, // D
                            float*       __restrict__ beta,      // 2
                            float invN)
{
    __shared__ float s0a[D], s1a[D];
    const int d = threadIdx.x;
    s0a[d] = sps[d]     * invN * att_inter[d];
    s1a[d] = sps[D + d] * invN * att_inter[d];
    __syncthreads();
    for (int off = D / 2; off > 0; off >>= 1) {
        if (d < off) { s0a[d] += s0a[d + off]; s1a[d] += s1a[d + off]; }
        __syncthreads();
    }
    if (d == 0) {
        const float a = s0a[0], b = s1a[0];
        const float m  = fmaxf(a, b);
        const float ea = __expf(a - m), eb = __expf(b - m);
        const float inv = 1.f / (ea + eb);
        beta[0] = ea * inv;
        beta[1] = eb * inv;
    }
}

__global__ void combine_kernel(const float* __restrict__ e1,
                               const float* __restrict__ e2,
                               const float* __restrict__ beta,
                               float*       __restrict__ out,
                               int n4)
{
    const int i = blockIdx.x * blockDim.x + threadIdx.x;
    if (i >= n4) return;
    const float b0 = beta[0], b1 = beta[1];
    const float4 v1 = ((const float4*)e1)[i];
    const float4 v2 = ((const float4*)e2)[i];
    float4 o;
    o.x = b0 * v1.x + b1 * v2.x;
    o.y = b0 * v1.y + b1 * v2.y;
    o.z = b0 * v1.z + b1 * v2.z;
    o.w = b0 * v1.w + b1 * v2.w;
    ((float4*)out)[i] = o;
}

// ---------------------------------------------------------------------------
extern "C" void kernel_launch(void* const* d_in, const int* in_sizes, int n_in,
                              void* d_out, int out_size, void* d_ws, size_t ws_size,
                              hipStream_t stream)
{
    const float* h_ref     = (const float*)d_in[0];
    const float* h1        = (const float*)d_in[1];
    const float* h2        = (const float*)d_in[2];
    const int*   nei1      = (const int*)  d_in[3];
    const int*   nei2      = (const int*)  d_in[4];
    const float* att1      = (const float*)d_in[5];
    const float* att2      = (const float*)d_in[6];
    const float* fcW       = (const float*)d_in[7];
    const float* fcb       = (const float*)d_in[8];
    const float* att_inter = (const float*)d_in[9];

    const int N  = in_sizes[0] / D;
    const int S1 = in_sizes[3] / N;
    const int S2 = in_sizes[4] / N;

    float* e1   = (float*)d_ws;
    float* e2   = e1 + (size_t)N * D;
    float* sps  = e2 + (size_t)N * D;   // 2*D column-sum accumulators
    float* beta = sps + 2 * D;          // 2 floats

    zero_kernel<<<1, 2 * D + 2, 0, stream>>>(sps, 2 * D + 2);

    const int nodeBlocks = (N + 3) / 4;
    intra_kernel<<<nodeBlocks, 128, 0, stream>>>(h_ref, h1, nei1, att1, e1, N, S1);
    intra_kernel<<<nodeBlocks, 128, 0, stream>>>(h_ref, h2, nei2, att2, e2, N, S2);

    fc_colsum_kernel<<<256, 256, 0, stream>>>(e1, fcW, fcb, sps,     N);
    fc_colsum_kernel<<<256, 256, 0, stream>>>(e2, fcW, fcb, sps + D, N);

    beta_kernel<<<1, D, 0, stream>>>(sps, att_inter, beta, 1.0f / (float)N);

    const int n4 = (N * D) / 4;
    combine_kernel<<<(n4 + 255) / 256, 256, 0, stream>>>(e1, e2, beta, (float*)d_out, n4);
}